// Autoencoder_90391881711665
// MI455X (gfx1250) — compile-verified
//
#include <hip/hip_runtime.h>

typedef __attribute__((ext_vector_type(2))) float v2f;
typedef __attribute__((ext_vector_type(8))) float v8f;

constexpr int KC = 512;          // num codes
constexpr int DC = 64;           // embedding dim
constexpr int NC = 131072;       // num vectors
constexpr int ROWS_PER_BLOCK = 128;
constexpr int NBLOCKS = NC / ROWS_PER_BLOCK;   // 1024
constexpr int THREADS = 256;

// ws layout (32-bit units): [0,512) enorm f32 | [512,1024) hist u32 | [1024,2048) loss partials f32
// out layout: [0] loss | [1, 1+N*D) quantized | [1+N*D] perplexity | [2+N*D, ...) encodings

__global__ void vq_prep(const float* __restrict__ emb, float* __restrict__ enorm,
                        unsigned* __restrict__ hist) {
    int g = blockIdx.x * blockDim.x + threadIdx.x;
    if (g < KC) {
        const float4* e4 = (const float4*)(emb + (size_t)g * DC);
        float s = 0.f;
#pragma unroll
        for (int i = 0; i < DC / 4; ++i) {
            float4 v = e4[i];
            s += v.x * v.x + v.y * v.y + v.z * v.z + v.w * v.w;
        }
        enorm[g] = s;
        hist[g] = 0u;
    }
}

__global__ __launch_bounds__(THREADS)
void vq_main(const float* __restrict__ x, const float* __restrict__ emb,
             const float* __restrict__ enorm, unsigned* __restrict__ hist,
             float* __restrict__ partials, float* __restrict__ out) {
    __shared__ int      sIdx[ROWS_PER_BLOCK];
    __shared__ unsigned sHist[KC];
    __shared__ float    sLoss[THREADS];

    const int t    = threadIdx.x;
    const int wave = t >> 5;
    const int lane = t & 31;
    const int half = lane >> 4;   // 0: lanes 0-15, 1: lanes 16-31
    const int lrow = lane & 15;

    sHist[t] = 0u;
    sHist[t + THREADS] = 0u;

    // ---------- Phase 1: WMMA scores + argmax, 16 rows per wave ----------
    const int rowBase = blockIdx.x * ROWS_PER_BLOCK + wave * 16;

    // A fragment (16x4 f32 per step): lane holds row (lane&15), K-pair 2*half..2*half+1
    v2f a[16];
    const v2f* ap = (const v2f*)(x + (size_t)(rowBase + lrow) * DC + 2 * half);
#pragma unroll
    for (int kk = 0; kk < 16; ++kk) a[kk] = ap[2 * kk];

    float bestv[8];
    int   besti[8];
#pragma unroll
    for (int r = 0; r < 8; ++r) { bestv[r] = -3.4e38f; besti[r] = 0; }

    for (int ct = 0; ct < KC / 16; ++ct) {
        const int code = ct * 16 + lrow;
        // B fragment (4x16 f32 per step): lane supplies column N=lrow, K-pair 2*half..+1
        const v2f* bp = (const v2f*)(emb + (size_t)code * DC + 2 * half);
        v8f acc = {};
#pragma unroll
        for (int kk = 0; kk < 16; ++kk) {
            v2f b = bp[2 * kk];
            acc = __builtin_amdgcn_wmma_f32_16x16x4_f32(
                /*neg_a=*/false, a[kk], /*neg_b=*/false, b,
                /*c_mod=*/(short)0, acc, /*reuse_a=*/false, /*reuse_b=*/false);
        }
        const float en = enorm[code];
#pragma unroll
        for (int r = 0; r < 8; ++r) {
            float s = 2.f * acc[r] - en;   // argmax of this == argmin of distance
            if (s > bestv[r]) { bestv[r] = s; besti[r] = code; }
        }
    }

    // reduce across the 16 lanes sharing each row (xor masks 1..8 stay in-half)
#pragma unroll
    for (int m = 1; m < 16; m <<= 1) {
#pragma unroll
        for (int r = 0; r < 8; ++r) {
            float ov = __shfl_xor(bestv[r], m, 32);
            int   oi = __shfl_xor(besti[r], m, 32);
            if (ov > bestv[r] || (ov == bestv[r] && oi < besti[r])) {
                bestv[r] = ov; besti[r] = oi;
            }
        }
    }
    if (lrow == 0) {
#pragma unroll
        for (int r = 0; r < 8; ++r) sIdx[wave * 16 + half * 8 + r] = besti[r];
    }
    __syncthreads();

    // ---------- Phase 2: histogram, quantized+loss, encodings ----------
    if (t < ROWS_PER_BLOCK) atomicAdd(&sHist[sIdx[t]], 1u);

    const int gRow0 = blockIdx.x * ROWS_PER_BLOCK;

    // quantized (out offset 1 -> only 4B aligned: scalar stores) + loss partial
    float lsum = 0.f;
#pragma unroll
    for (int i = 0; i < 8; ++i) {
        int linear = t + THREADS * i;          // 0..2047 (= 128 rows * 16 segs)
        int row = linear >> 4;
        int seg = linear & 15;
        int idx = sIdx[row];
        float4 q  = ((const float4*)(emb + (size_t)idx * DC))[seg];
        float4 xi = ((const float4*)(x + (size_t)(gRow0 + row) * DC))[seg];
        float dx = q.x - xi.x, dy = q.y - xi.y, dz = q.z - xi.z, dw = q.w - xi.w;
        lsum += dx * dx + dy * dy + dz * dz + dw * dw;
        size_t o = 1 + (size_t)(gRow0 + row) * DC + (size_t)seg * 4;
        out[o + 0] = q.x; out[o + 1] = q.y; out[o + 2] = q.z; out[o + 3] = q.w;
    }

    // encodings: base offset 2+N*D floats is 8B aligned -> float2 stores, fully coalesced
    float2* ebase = (float2*)(out + 2 + (size_t)NC * DC);
#pragma unroll 4
    for (int row = 0; row < ROWS_PER_BLOCK; ++row) {
        int idx = sIdx[row];
        float2 v;
        v.x = (idx == 2 * t)     ? 1.f : 0.f;
        v.y = (idx == 2 * t + 1) ? 1.f : 0.f;
        ebase[(size_t)(gRow0 + row) * (KC / 2) + t] = v;
    }

    // block loss reduction (fixed tree -> deterministic)
    sLoss[t] = lsum;
    __syncthreads();
    for (int s = THREADS / 2; s > 0; s >>= 1) {
        if (t < s) sLoss[t] += sLoss[t + s];
        __syncthreads();
    }
    if (t == 0) partials[blockIdx.x] = sLoss[0];

    // flush histogram (integer atomics -> deterministic)
    unsigned c0 = sHist[t];
    unsigned c1 = sHist[t + THREADS];
    if (c0) atomicAdd(&hist[t], c0);
    if (c1) atomicAdd(&hist[t + THREADS], c1);
}

__global__ __launch_bounds__(THREADS)
void vq_final(const unsigned* __restrict__ hist, const float* __restrict__ partials,
              float* __restrict__ out) {
    __shared__ float sA[THREADS], sB[THREADS];
    int t = threadIdx.x;
    float ls = partials[t] + partials[t + 256] + partials[t + 512] + partials[t + 768];
    float es = 0.f;
#pragma unroll
    for (int j = 0; j < 2; ++j) {
        float p = (float)hist[t + j * 256] * (1.f / (float)NC);
        es += p * __logf(p + 1e-10f);
    }
    sA[t] = ls; sB[t] = es;
    __syncthreads();
    for (int s = THREADS / 2; s > 0; s >>= 1) {
        if (t < s) { sA[t] += sA[t + s]; sB[t] += sB[t + s]; }
        __syncthreads();
    }
    if (t == 0) {
        out[0] = 1.25f * sA[0] / (float)((size_t)NC * DC);   // q_loss + 0.25*e_loss
        out[1 + (size_t)NC * DC] = __expf(-sB[0]);           // perplexity
    }
}

extern "C" void kernel_launch(void* const* d_in, const int* in_sizes, int n_in,
                              void* d_out, int out_size, void* d_ws, size_t ws_size,
                              hipStream_t stream) {
    (void)in_sizes; (void)n_in; (void)out_size; (void)ws_size;
    const float* x   = (const float*)d_in[0];
    const float* emb = (const float*)d_in[1];
    float*    out      = (float*)d_out;
    float*    enorm    = (float*)d_ws;
    unsigned* hist     = (unsigned*)d_ws + 512;
    float*    partials = (float*)d_ws + 1024;

    vq_prep<<<2, THREADS, 0, stream>>>(emb, enorm, hist);
    vq_main<<<NBLOCKS, THREADS, 0, stream>>>(x, emb, enorm, hist, partials, out);
    vq_final<<<1, THREADS, 0, stream>>>(hist, partials, out);
}